// MOLAttn_38740605009938
// MI455X (gfx1250) — compile-verified
//
#include <hip/hip_runtime.h>
#include <hip/hip_bf16.h>
#include <math.h>

// ---------------- problem constants ----------------
#define BB 64
#define NN 2048
#define EE 512
#define DD 256
#define AA 1024
#define KK 5
#define GG (3 * AA)       // 3072
#define KIH (EE + DD)     // 768, att_in width

// ---------------- workspace layout (float offsets) ----------------
#define OFF_ATT   0                         // [B, 768]
#define OFF_GI    (OFF_ATT  + BB*KIH)       // [B, 3072]
#define OFF_GH    (OFF_GI   + BB*GG)        // [B, 3072]
#define OFF_H     (OFF_GH   + BB*GG)        // [B, A]
#define OFF_T     (OFF_H    + BB*AA)        // [K, B, A]
#define OFF_P     (OFF_T    + KK*BB*AA)     // [K, B, 3]
#define OFF_MEAN  (OFF_P    + KK*BB*3)      // [K, B]
#define OFF_SCALE (OFF_MEAN + KK*BB)        // [K, B]
#define OFF_W     (OFF_SCALE+ KK*BB)        // [K, B]
#define OFF_A     (OFF_W    + KK*BB)        // [B, N]
#define NSPLIT 16
#define OFF_PART  (OFF_A    + BB*NN)        // [NSPLIT, B, E]
#define WS_FLOATS (OFF_PART + NSPLIT*BB*EE) // ~6 MB total

// d_out layout: c [B*E] | h [B*A] | w [K*B]
#define OUT_C 0
#define OUT_H (BB*EE)
#define OUT_W (BB*EE + BB*AA)

typedef float  v2f __attribute__((ext_vector_type(2)));
typedef float  v8f __attribute__((ext_vector_type(8)));
typedef float  f4v __attribute__((ext_vector_type(4)));

__device__ __forceinline__ float sigf(float x) { return 1.0f / (1.0f + expf(-x)); }

// ---------------- K0: att_in = concat(dec_z, c0) ----------------
__global__ void k_build_attin(const float* __restrict__ dec_z,
                              const float* __restrict__ c0,
                              float* __restrict__ ws) {
    int idx = blockIdx.x * blockDim.x + threadIdx.x;       // over B*768
    if (idx >= BB * KIH) return;
    int b = idx / KIH, c = idx % KIH;
    float v = (c < DD) ? dec_z[b * DD + c] : c0[b * EE + (c - DD)];
    ws[OFF_ATT + idx] = v;
}

// ---------------- K1: gate GEMMs via V_WMMA_F32_16X16X4_F32 ----------------
// OUT[m][n] = sum_k X[m][k] * W[n][k] + bias[n]   (X @ W^T)
// z==0: OUT=gi, X=att_in (ld 768),  W=W_ih, bias=b_ih
// z==1: OUT=gh, X=h0     (ld 1024), W=W_hh, bias=b_hh
__global__ __launch_bounds__(32) void k_gemm_gates(const float* __restrict__ W_ih,
                                                   const float* __restrict__ W_hh,
                                                   const float* __restrict__ b_ih,
                                                   const float* __restrict__ b_hh,
                                                   const float* __restrict__ h0,
                                                   float* __restrict__ ws) {
    const int l  = threadIdx.x & 15;
    const int hf = threadIdx.x >> 4;
    const int n0 = blockIdx.x * 16;
    const int m0 = blockIdx.y * 16;
    const int job = blockIdx.z;

    const float* X    = (job == 0) ? (ws + OFF_ATT) : h0;
    const float* W    = (job == 0) ? W_ih : W_hh;
    const float* bias = (job == 0) ? b_ih : b_hh;
    float*       OUT  = (job == 0) ? (ws + OFF_GI) : (ws + OFF_GH);
    const int    Kd   = (job == 0) ? KIH : AA;

    v8f acc = {};
    const float* xrow = X + (size_t)(m0 + l) * Kd + 2 * hf;
    const float* wrow = W + (size_t)(n0 + l) * Kd + 2 * hf;
#pragma unroll 4
    for (int k0 = 0; k0 < Kd; k0 += 4) {
        v2f a, b;
        a.x = xrow[k0];  a.y = xrow[k0 + 1];
        b.x = wrow[k0];  b.y = wrow[k0 + 1];
        acc = __builtin_amdgcn_wmma_f32_16x16x4_f32(false, a, false, b,
                                                    (short)0, acc, false, false);
    }
    const float bv = bias[n0 + l];
#pragma unroll
    for (int v = 0; v < 8; ++v) {
        int row = m0 + v + 8 * hf;
        OUT[(size_t)row * GG + n0 + l] = acc[v] + bv;
    }
}

// ---------------- K2: GRU gate combine -> h ----------------
__global__ void k_gru_combine(const float* __restrict__ h0,
                              float* __restrict__ ws,
                              float* __restrict__ out_h) {
    int idx = blockIdx.x * blockDim.x + threadIdx.x;       // over B*A
    if (idx >= BB * AA) return;
    int b = idx / AA, j = idx % AA;
    const float* gi = ws + OFF_GI + (size_t)b * GG;
    const float* gh = ws + OFF_GH + (size_t)b * GG;
    float r  = sigf(gi[j]          + gh[j]);
    float z  = sigf(gi[j + AA]     + gh[j + AA]);
    float nn = tanhf(gi[j + 2*AA]  + r * gh[j + 2*AA]);
    float hv = (1.0f - z) * nn + z * h0[idx];
    ws[OFF_H + idx] = hv;
    out_h[idx]      = hv;     // output h[None] region
}

// ---------------- K3: t[k] = tanh(h @ W1[k] + b1[k]) via WMMA ----------------
// t[k,b,c] = sum_a h[b,a] * W1[k,a,c]   (B-matrix not transposed)
__global__ __launch_bounds__(32) void k_gemm_mlp1(const float* __restrict__ W1,
                                                  const float* __restrict__ b1,
                                                  float* __restrict__ ws) {
    const int l  = threadIdx.x & 15;
    const int hf = threadIdx.x >> 4;
    const int n0 = blockIdx.x * 16;
    const int m0 = blockIdx.y * 16;
    const int k  = blockIdx.z;

    const float* H   = ws + OFF_H;
    const float* W1k = W1 + (size_t)k * AA * AA;

    v8f acc = {};
    const float* xrow = H + (size_t)(m0 + l) * AA + 2 * hf;
#pragma unroll 4
    for (int k0 = 0; k0 < AA; k0 += 4) {
        v2f a, b;
        a.x = xrow[k0];  a.y = xrow[k0 + 1];
        const float* wp = W1k + (size_t)(k0 + 2 * hf) * AA + n0 + l;
        b.x = wp[0];     b.y = wp[AA];
        acc = __builtin_amdgcn_wmma_f32_16x16x4_f32(false, a, false, b,
                                                    (short)0, acc, false, false);
    }
    const float bv = b1[k * AA + n0 + l];
    float* T = ws + OFF_T + (size_t)k * BB * AA;
#pragma unroll
    for (int v = 0; v < 8; ++v) {
        int row = m0 + v + 8 * hf;
        T[(size_t)row * AA + n0 + l] = tanhf(acc[v] + bv);
    }
}

// ---------------- K4: p[k,b,:] = t[k,b,:] @ W2[k] + b2[k] ----------------
__global__ __launch_bounds__(256) void k_mlp2(const float* __restrict__ W2,
                                              const float* __restrict__ b2,
                                              float* __restrict__ ws) {
    const int kb = blockIdx.x;            // K*B blocks
    const int k = kb / BB, b = kb % BB;
    const float* T   = ws + OFF_T + ((size_t)k * BB + b) * AA;
    const float* W2k = W2 + (size_t)k * AA * 3;
    float s0 = 0.f, s1 = 0.f, s2 = 0.f;
    for (int c = threadIdx.x; c < AA; c += 256) {
        float tv = T[c];
        s0 += tv * W2k[c * 3 + 0];
        s1 += tv * W2k[c * 3 + 1];
        s2 += tv * W2k[c * 3 + 2];
    }
    __shared__ float red[3][256];
    red[0][threadIdx.x] = s0; red[1][threadIdx.x] = s1; red[2][threadIdx.x] = s2;
    __syncthreads();
    for (int off = 128; off > 0; off >>= 1) {
        if (threadIdx.x < off) {
            red[0][threadIdx.x] += red[0][threadIdx.x + off];
            red[1][threadIdx.x] += red[1][threadIdx.x + off];
            red[2][threadIdx.x] += red[2][threadIdx.x + off];
        }
        __syncthreads();
    }
    if (threadIdx.x == 0) {
        float* P = ws + OFF_P + (size_t)kb * 3;
        P[0] = red[0][0] + b2[k * 3 + 0];
        P[1] = red[1][0] + b2[k * 3 + 1];
        P[2] = red[2][0] + b2[k * 3 + 2];
    }
}

// ---------------- K5: means/scales + softmax over K -> w ----------------
__global__ void k_mix_params(float* __restrict__ ws, float* __restrict__ out_w) {
    int b = threadIdx.x;                  // 64 threads
    if (b >= BB) return;
    float logits[KK];
    float mx = -1e30f;
    for (int k = 0; k < KK; ++k) {
        const float* P = ws + OFF_P + ((size_t)k * BB + b) * 3;
        ws[OFF_MEAN  + k * BB + b] = expf(P[0]);
        ws[OFF_SCALE + k * BB + b] = expf(P[1]);
        logits[k] = P[2];
        mx = fmaxf(mx, P[2]);
    }
    float denom = 0.f;
    float e[KK];
    for (int k = 0; k < KK; ++k) { e[k] = expf(logits[k] - mx); denom += e[k]; }
    for (int k = 0; k < KK; ++k) {
        float wv = e[k] / denom;
        ws[OFF_W + k * BB + b] = wv;
        out_w[k * BB + b]      = wv;      // output w region [K,B]
    }
}

// ---------------- K6: attention weights a[b,n] ----------------
__global__ __launch_bounds__(256) void k_attn_weights(float* __restrict__ ws) {
    int b = blockIdx.x;
    int n = blockIdx.y * 256 + threadIdx.x;
    float x  = (float)n;
    float av = 0.f;
    for (int k = 0; k < KK; ++k) {
        float m = ws[OFF_MEAN  + k * BB + b];
        float s = ws[OFF_SCALE + k * BB + b];
        float w = ws[OFF_W     + k * BB + b];
        float L1 = 1.0f / (1.0f + expf((x + 0.5f - m) / s));
        float L2 = 1.0f / (1.0f + expf((x - 0.5f - m) / s));
        av += w * (L1 - L2);              // the +1e-6 terms cancel exactly
    }
    ws[OFF_A + (size_t)b * NN + n] = av;
}

// ---------------- K7: streaming partial context (NT b128 loads) ----------
// 128 threads * float4 = one full 2KB row of enc_z per iteration, coalesced.
// enc_z (256 MB) > L2 (192 MB) and is read exactly once -> non-temporal.
__global__ __launch_bounds__(128) void k_context_partial(const float* __restrict__ enc_z,
                                                         float* __restrict__ ws) {
    const int b  = blockIdx.x;
    const int ns = blockIdx.y;            // NSPLIT splits along N
    const int rows = NN / NSPLIT;         // 128
    const int nbeg = ns * rows;
    f4v acc = {0.f, 0.f, 0.f, 0.f};
    const float* arow = ws + OFF_A + (size_t)b * NN;
    const f4v* base = (const f4v*)(enc_z + ((size_t)b * NN + nbeg) * EE) + threadIdx.x;
#pragma unroll 2
    for (int r = 0; r < rows; ++r) {
        float s = arow[nbeg + r];         // uniform across block -> scalar load
        f4v v = __builtin_nontemporal_load(base + (size_t)r * (EE / 4));
        acc.x += s * v.x;
        acc.y += s * v.y;
        acc.z += s * v.z;
        acc.w += s * v.w;
    }
    f4v* part = (f4v*)(ws + OFF_PART + ((size_t)ns * BB + b) * EE);
    part[threadIdx.x] = acc;
}

// ---------------- K8: deterministic reduce of partials -> c ----------------
__global__ __launch_bounds__(256) void k_context_reduce(float* __restrict__ ws,
                                                        float* __restrict__ out_c) {
    int idx = blockIdx.x * 256 + threadIdx.x;   // over B*E
    if (idx >= BB * EE) return;
    int b = idx / EE, e = idx % EE;
    float s = 0.f;
    for (int ns = 0; ns < NSPLIT; ++ns)
        s += ws[OFF_PART + ((size_t)ns * BB + b) * EE + e];
    out_c[idx] = s;                             // c[None] region (offset 0)
}

// ---------------- launch ----------------
extern "C" void kernel_launch(void* const* d_in, const int* in_sizes, int n_in,
                              void* d_out, int out_size, void* d_ws, size_t ws_size,
                              hipStream_t stream) {
    const float* enc_z = (const float*)d_in[0];
    const float* dec_z = (const float*)d_in[1];
    const float* h0    = (const float*)d_in[2];
    const float* c0    = (const float*)d_in[3];
    const float* W_ih  = (const float*)d_in[4];
    const float* W_hh  = (const float*)d_in[5];
    const float* b_ih  = (const float*)d_in[6];
    const float* b_hh  = (const float*)d_in[7];
    const float* W1    = (const float*)d_in[8];
    const float* b1    = (const float*)d_in[9];
    const float* W2    = (const float*)d_in[10];
    const float* b2    = (const float*)d_in[11];
    (void)in_sizes; (void)n_in; (void)out_size; (void)ws_size;

    float* ws  = (float*)d_ws;
    float* out = (float*)d_out;

    // K0: att_in
    k_build_attin<<<dim3((BB * KIH + 255) / 256), dim3(256), 0, stream>>>(dec_z, c0, ws);
    // K1: gi/gh GEMMs (WMMA f32 16x16x4). grid: 3072/16 x 64/16 x 2 jobs
    k_gemm_gates<<<dim3(GG / 16, BB / 16, 2), dim3(32), 0, stream>>>(
        W_ih, W_hh, b_ih, b_hh, h0, ws);
    // K2: gate combine -> h
    k_gru_combine<<<dim3((BB * AA + 255) / 256), dim3(256), 0, stream>>>(
        h0, ws, out + OUT_H);
    // K3: t = tanh(h @ W1 + b1) (WMMA). grid: 1024/16 x 64/16 x K
    k_gemm_mlp1<<<dim3(AA / 16, BB / 16, KK), dim3(32), 0, stream>>>(W1, b1, ws);
    // K4: p = t @ W2 + b2
    k_mlp2<<<dim3(KK * BB), dim3(256), 0, stream>>>(W2, b2, ws);
    // K5: mixture params + softmax -> w
    k_mix_params<<<dim3(1), dim3(64), 0, stream>>>(ws, out + OUT_W);
    // K6: attention weights a[b,n]
    k_attn_weights<<<dim3(BB, NN / 256), dim3(256), 0, stream>>>(ws);
    // K7: streaming context partials (the 256 MB / 23.3 TB/s critical path)
    k_context_partial<<<dim3(BB, NSPLIT), dim3(128), 0, stream>>>(enc_z, ws);
    // K8: deterministic final reduce -> c
    k_context_reduce<<<dim3((BB * EE + 255) / 256), dim3(256), 0, stream>>>(
        ws, out + OUT_C);
}